// GRNN_3547642986518
// MI455X (gfx1250) — compile-verified
//
#include <hip/hip_runtime.h>
#include <stdint.h>

// Sequential Kalman/SDE scan, T = 262144 steps.
//   K1: two independent serial chains (Riccati cov / cubic f) on separate
//       single-lane waves -> compiler maps them onto the SALU float pipe
//       (s_fmac_f32, ~2-3 cycle dependent latency). Issue-bound, so ops/step
//       are minimized via Horner grouping; all foldable multiplies moved to
//       the parallel consumers. This phase is the irreducible
//       serial dependency chain.
//   K2: per-chunk composition of the affine x-map  x' = M_n x + b_n.
//   K3: serial scan over 2048 chunk transforms (negligible; SALU-vectorized
//       by the compiler into s_load_b512 + s_fmac chains).
//   K4: parallel replay, dy_hats = [c*dt*x0, 0]; coefficients double-buffered
//       through LDS with global_load_async_to_lds + s_wait_asynccnt (ASYNCcnt).
//
// Layouts: TR=true  -> coefficients transposed [step-in-chunk][chunk] so K2/K4
//                      reads are coalesced across the 256 threads of a block.
//          TR=false -> flat [n]; fallback when ws is tiny: covk borrows the
//                      dy_hats region of d_out (read-before-overwrite safe).

#define T_STEPS 262144
#define CHUNK   128
#define LOG_CH  7
#define NCH     (T_STEPS / CHUNK)   // 2048
#define DT_F    0.001f
#define DELAY_F 1.0f
#define TI      4                   // async staging tile (steps)

static_assert(CHUNK == (1 << LOG_CH), "chunk mismatch");

// ---------------------------------------------------------------- K1
template <bool TR>
__global__ __launch_bounds__(32, 1) void k1_serial_chains(
    const float* __restrict__ state0, const float* __restrict__ f0v,
    const float* __restrict__ K0,  const float* __restrict__ K1m,
    const float* __restrict__ K2m, const float* __restrict__ K2c,
    const float* __restrict__ K3m, const float* __restrict__ K312,
    const float* __restrict__ K321, const float* __restrict__ Am,
    const float* __restrict__ Cm,  const float* __restrict__ Dm,
    float2* __restrict__ covk, float* __restrict__ fdt,
    float* __restrict__ out_tail)
{
    if (threadIdx.x != 0) return;
    asm volatile("s_setprio 3");
    const int stp = TR ? NCH : 1;

    if (blockIdx.x == 0) {
        // ---- covariance Riccati chain (raw (vx,cxp) stored; c folded later)
        float vx = state0[2], vp = state0[3], cxp = state0[4], t = state0[5];
        const float a00 = Am[0], a01 = Am[1], a10 = Am[2], a11 = Am[3];
        const float c   = Cm[0];
        const float c2  = c * c;
        const float d00 = Dm[0], d11 = Dm[3];
        const float ta00 = 2.0f * a00, ta01 = 2.0f * a01;
        const float ta10 = 2.0f * a10, ta11 = 2.0f * a11;
        const float asum = a00 + a11;
        for (int ch = 0; ch < NCH; ++ch) {
            float2* pk = covk + (TR ? ch : ch * CHUNK);
#pragma unroll 4
            for (int i = 0; i < CHUNK; ++i, pk += stp) {
                *pk = make_float2(vx, cxp);
                // vx' = vx + dt*( vx*(2a00 - c2*vx) + (2a01*cxp + d00) )
                float i1  = fmaf(-c2, vx, ta00);
                float rvx = fmaf(vx, i1, fmaf(ta01, cxp, d00));
                // vp' = vp + dt*( cxp*(2a10 - c2*cxp) + (2a11*vp + d11) )
                float i2  = fmaf(-c2, cxp, ta10);
                float rvp = fmaf(cxp, i2, fmaf(ta11, vp, d11));
                // cxp' = cxp + dt*( vx*(a10 - c2*cxp) + (a00+a11)*cxp + a01*vp )
                float i3  = fmaf(-c2, cxp, a10);
                float rcx = fmaf(vx, i3, fmaf(asum, cxp, a01 * vp));
                vx  = fmaf(DT_F, rvx, vx);
                vp  = fmaf(DT_F, rvp, vp);
                cxp = fmaf(DT_F, rcx, cxp);
                t  += DT_F;
            }
        }
        out_tail[2] = vx; out_tail[3] = vp; out_tail[4] = cxp; out_tail[5] = t;
    } else {
        // ---- cubic f chain (raw fnew0 stored; projF10*dt folded later) ----
        float fa = f0v[0], fb = f0v[1];
        const float k00 = K0[0], k01 = K0[1];
        const float l00 = K1m[0], l01 = K1m[1], l10 = K1m[2], l11 = K1m[3];
        const float q00 = K2m[0], q01 = K2m[1], q10 = K2m[2], q11 = K2m[3];
        const float r0 = K2c[0] + K2c[1], r1 = K2c[2] + K2c[3];  // f*ffl=(p,p)
        const float w00 = K3m[0], w01 = K3m[1], w10 = K3m[2], w11 = K3m[3];
        // f*ffl^2 = (fa*s1, fb*s0), f^2*ffl = (s0*fb, s1*fa): merged coeffs
        const float cc0 = K312[0] + K321[1], cd0 = K312[1] + K321[0];
        const float cc1 = K312[2] + K321[3], cd1 = K312[3] + K321[2];
        const float ddt = DELAY_F * DT_F;
        for (int ch = 0; ch < NCH; ++ch) {
            float* pf = fdt + (TR ? ch : ch * CHUNK);
#pragma unroll 4
            for (int i = 0; i < CHUNK; ++i, pf += stp) {
                float s0 = fa * fa, s1 = fb * fb, p = fa * fb;
                // df0/dt = base0 + s0*(q00 + w00*fa + cd0*fb)
                //               + s1*(q01 + cc0*fa + w01*fb)
                float G0 = fmaf(cd0, fb, fmaf(w00, fa, q00));
                float H0 = fmaf(w01, fb, fmaf(cc0, fa, q01));
                float b0 = fmaf(r0, p, fmaf(l01, fb, fmaf(l00, fa, k00)));
                float a0 = fmaf(s1, H0, fmaf(s0, G0, b0));
                float G1 = fmaf(cd1, fb, fmaf(w10, fa, q10));
                float H1 = fmaf(w11, fb, fmaf(cc1, fa, q11));
                float b1 = fmaf(r1, p, fmaf(l11, fb, fmaf(l10, fa, k01)));
                float a1 = fmaf(s1, H1, fmaf(s0, G1, b1));
                float fna = fmaf(ddt, a0, fa);   // fnew
                float fnb = fmaf(ddt, a1, fb);
                *pf = fna;
                fa = fna; fb = fnb;
            }
        }
        out_tail[6] = fa; out_tail[7] = fb;
    }
}

// ---------------------------------------------------------------- K2
template <bool TR>
__global__ void k2_chunk_transform(
    const float* __restrict__ dy, const float2* __restrict__ covk,
    const float* __restrict__ fdt, const float* __restrict__ Am,
    const float* __restrict__ Cm, const float* __restrict__ PF,
    float* __restrict__ chT)
{
    int c = blockIdx.x * blockDim.x + threadIdx.x;
    if (c >= NCH) return;
    const float a00 = Am[0], a01 = Am[1], a10 = Am[2], a11 = Am[3];
    const float e00 = fmaf(DT_F, a00, 1.0f), e01 = DT_F * a01;
    const float e10 = DT_F * a10, e11 = fmaf(DT_F, a11, 1.0f);
    const float cc  = Cm[0];
    const float dcc = DT_F * cc * cc;
    const float pzdt = PF[2] * DT_F;
    float M00 = 1.f, M01 = 0.f, M10 = 0.f, M11 = 1.f, B0 = 0.f, B1 = 0.f;
    const int base = c * CHUNK;
    for (int i = 0; i < CHUNK; ++i) {
        const int n = base + i;
        const int q = TR ? (i * NCH + c) : n;
        float2 k  = covk[q];
        float fd  = fdt[q];
        float dy0 = dy[2 * n];
        __builtin_prefetch((const char*)(covk + q) + (TR ? NCH * 32 : 256), 0, 1);
        float m00 = fmaf(-dcc, k.x, e00);
        float m10 = fmaf(-dcc, k.y, e10);
        float b0  = (cc * k.x) * dy0;
        float b1  = fmaf(cc * k.y, dy0, fd * pzdt);
        float nM00 = fmaf(m00, M00, e01 * M10);
        float nM01 = fmaf(m00, M01, e01 * M11);
        float nM10 = fmaf(m10, M00, e11 * M10);
        float nM11 = fmaf(m10, M01, e11 * M11);
        float nB0  = fmaf(m00, B0, fmaf(e01, B1, b0));
        float nB1  = fmaf(m10, B0, fmaf(e11, B1, b1));
        M00 = nM00; M01 = nM01; M10 = nM10; M11 = nM11; B0 = nB0; B1 = nB1;
    }
    float* o = chT + 6 * c;
    o[0] = M00; o[1] = M01; o[2] = M10; o[3] = M11; o[4] = B0; o[5] = B1;
}

// ---------------------------------------------------------------- K3
__global__ void k3_chunk_scan(
    const float* __restrict__ state0, const float* __restrict__ chT,
    float2* __restrict__ xs, float* __restrict__ out_tail)
{
    if (threadIdx.x != 0 || blockIdx.x != 0) return;
    asm volatile("s_setprio 3");
    float x0 = state0[0], x1 = state0[1];
    for (int c = 0; c < NCH; ++c) {
        xs[c] = make_float2(x0, x1);
        const float* t = chT + 6 * c;
        float nx0 = fmaf(t[0], x0, fmaf(t[1], x1, t[4]));
        float nx1 = fmaf(t[2], x0, fmaf(t[3], x1, t[5]));
        x0 = nx0; x1 = nx1;
    }
    out_tail[0] = x0; out_tail[1] = x1;
}

// ---------------------------------------------------------------- K4 (fallback)
// covk may alias `out`; per-step load precedes the store -> no __restrict__.
__global__ void k4_replay_plain(
    const float* __restrict__ dy, const float2* covk,
    const float* __restrict__ fdt, const float2* __restrict__ xs,
    const float* __restrict__ Am, const float* __restrict__ Cm,
    const float* __restrict__ PF, float2* out)
{
    int c = blockIdx.x * blockDim.x + threadIdx.x;
    if (c >= NCH) return;
    const float a00 = Am[0], a01 = Am[1], a10 = Am[2], a11 = Am[3];
    const float e00 = fmaf(DT_F, a00, 1.0f), e01 = DT_F * a01;
    const float e10 = DT_F * a10, e11 = fmaf(DT_F, a11, 1.0f);
    const float cc  = Cm[0];
    const float dcc = DT_F * cc * cc;
    const float cdt = cc * DT_F;
    const float pzdt = PF[2] * DT_F;
    float2 x = xs[c];
    float x0 = x.x, x1 = x.y;
    const int base = c * CHUNK;
    for (int i = 0; i < CHUNK; ++i) {
        const int n = base + i;
        float2 k  = covk[n];
        float dy0 = dy[2 * n];
        float fd  = fdt[n];
        __builtin_prefetch((const char*)(covk + n) + 256, 0, 1);
        out[n] = make_float2(cdt * x0, 0.0f);
        float m00 = fmaf(-dcc, k.x, e00);
        float m10 = fmaf(-dcc, k.y, e10);
        float nx0 = fmaf(m00, x0, fmaf(e01, x1, (cc * k.x) * dy0));
        float nx1 = fmaf(m10, x0, fmaf(e11, x1, fmaf(cc * k.y, dy0, fd * pzdt)));
        x0 = nx0; x1 = nx1;
    }
}

// ---------------------------------------------------------------- K4 (async)
// Transposed coefficients, double-buffered through LDS with CDNA5 async
// global->LDS loads. Each lane stages only its own elements, so only the
// per-wave s_wait_asynccnt is needed (no block barrier).
__global__ void k4_replay_async(
    const float* __restrict__ dy, const float2* __restrict__ covk_t,
    const float* __restrict__ fdt_t, const float2* __restrict__ xs,
    const float* __restrict__ Am, const float* __restrict__ Cm,
    const float* __restrict__ PF, float2* __restrict__ out)
{
    __shared__ float2 sk[2][TI * 256];
    __shared__ float  sf[2][TI * 256];
    __shared__ float  sd[2][TI * 256];
    const int tid = threadIdx.x;
    const int c   = blockIdx.x * 256 + tid;
    const float a00 = Am[0], a01 = Am[1], a10 = Am[2], a11 = Am[3];
    const float e00 = fmaf(DT_F, a00, 1.0f), e01 = DT_F * a01;
    const float e10 = DT_F * a10, e11 = fmaf(DT_F, a11, 1.0f);
    const float cc  = Cm[0];
    const float dcc = DT_F * cc * cc;
    const float cdt = cc * DT_F;
    const float pzdt = PF[2] * DT_F;
    float2 x = xs[c];
    float x0 = x.x, x1 = x.y;

    auto issue = [&](int buf, int t0) {
#pragma unroll
        for (int j = 0; j < TI; ++j) {
            const int i = t0 + j;
            unsigned lk = (unsigned)(uintptr_t)&sk[buf][j * 256 + tid];
            unsigned lf = (unsigned)(uintptr_t)&sf[buf][j * 256 + tid];
            unsigned ld = (unsigned)(uintptr_t)&sd[buf][j * 256 + tid];
            unsigned gk = (unsigned)((i * NCH + c) * sizeof(float2));
            unsigned gf = (unsigned)((i * NCH + c) * sizeof(float));
            unsigned gd = (unsigned)((c * CHUNK + i) * 2u * sizeof(float));
            asm volatile("global_load_async_to_lds_b64 %0, %1, %2"
                         :: "v"(lk), "v"(gk), "s"((const void*)covk_t) : "memory");
            asm volatile("global_load_async_to_lds_b32 %0, %1, %2"
                         :: "v"(lf), "v"(gf), "s"((const void*)fdt_t) : "memory");
            asm volatile("global_load_async_to_lds_b32 %0, %1, %2"
                         :: "v"(ld), "v"(gd), "s"((const void*)dy) : "memory");
        }
    };

    issue(0, 0);
    int buf = 0;
    for (int t0 = 0; t0 < CHUNK; t0 += TI) {
        asm volatile("s_wait_asynccnt 0x0" ::: "memory");
        if (t0 + TI < CHUNK) issue(buf ^ 1, t0 + TI);
#pragma unroll
        for (int j = 0; j < TI; ++j) {
            float2 k  = sk[buf][j * 256 + tid];
            float fd  = sf[buf][j * 256 + tid];
            float dy0 = sd[buf][j * 256 + tid];
            const int n = c * CHUNK + t0 + j;
            out[n] = make_float2(cdt * x0, 0.0f);
            float m00 = fmaf(-dcc, k.x, e00);
            float m10 = fmaf(-dcc, k.y, e10);
            float nx0 = fmaf(m00, x0, fmaf(e01, x1, (cc * k.x) * dy0));
            float nx1 = fmaf(m10, x0,
                             fmaf(e11, x1, fmaf(cc * k.y, dy0, fd * pzdt)));
            x0 = nx0; x1 = nx1;
        }
        buf ^= 1;
    }
}

// ---------------------------------------------------------------- host
extern "C" void kernel_launch(void* const* d_in, const int* in_sizes, int n_in,
                              void* d_out, int out_size, void* d_ws, size_t ws_size,
                              hipStream_t stream)
{
    (void)in_sizes; (void)n_in; (void)out_size;
    const float* dy     = (const float*)d_in[0];
    const float* state0 = (const float*)d_in[1];
    const float* f0v    = (const float*)d_in[2];
    const float* K0     = (const float*)d_in[3];
    const float* K1m    = (const float*)d_in[4];
    const float* K2m    = (const float*)d_in[5];
    const float* K2c    = (const float*)d_in[6];
    const float* K3m    = (const float*)d_in[7];
    const float* K312   = (const float*)d_in[8];
    const float* K321   = (const float*)d_in[9];
    const float* Am     = (const float*)d_in[10];
    const float* Cm     = (const float*)d_in[11];
    const float* Dm     = (const float*)d_in[12];
    const float* PF     = (const float*)d_in[13];
    float* out = (float*)d_out;

    char* w = (char*)d_ws;
    float*  chT = (float*)w;   w += sizeof(float)  * 6 * NCH;
    float2* xs  = (float2*)w;  w += sizeof(float2) * NCH;
    float*  fdt = (float*)w;   w += sizeof(float)  * T_STEPS;
    size_t used = (size_t)(w - (char*)d_ws);
    const bool big = (ws_size >= used + sizeof(float2) * T_STEPS);

    if (big) {
        float2* covk = (float2*)w;
        k1_serial_chains<true><<<2, 32, 0, stream>>>(state0, f0v, K0, K1m, K2m,
            K2c, K3m, K312, K321, Am, Cm, Dm, covk, fdt, out + 2 * T_STEPS);
        k2_chunk_transform<true><<<NCH / 256, 256, 0, stream>>>(dy, covk, fdt,
            Am, Cm, PF, chT);
        k3_chunk_scan<<<1, 32, 0, stream>>>(state0, chT, xs, out + 2 * T_STEPS);
        k4_replay_async<<<NCH / 256, 256, 0, stream>>>(dy, covk, fdt, xs,
            Am, Cm, PF, (float2*)out);
    } else {
        float2* covk = (float2*)out;   // borrow dy_hats region (flat layout)
        k1_serial_chains<false><<<2, 32, 0, stream>>>(state0, f0v, K0, K1m, K2m,
            K2c, K3m, K312, K321, Am, Cm, Dm, covk, fdt, out + 2 * T_STEPS);
        k2_chunk_transform<false><<<NCH / 256, 256, 0, stream>>>(dy, covk, fdt,
            Am, Cm, PF, chT);
        k3_chunk_scan<<<1, 32, 0, stream>>>(state0, chT, xs, out + 2 * T_STEPS);
        k4_replay_plain<<<NCH / 256, 256, 0, stream>>>(dy, covk, fdt, xs,
            Am, Cm, PF, (float2*)out);
    }
}